// mymodel_13915694039600
// MI455X (gfx1250) — compile-verified
//
#include <hip/hip_runtime.h>
#include <cstddef>
#include <cstdint>

typedef float v2f __attribute__((ext_vector_type(2)));
typedef float v8f __attribute__((ext_vector_type(8)));

namespace {
constexpr int cB  = 4096;
constexpr int cF  = 50;
constexpr int cDU = 128;
constexpr int cDI = 256;
constexpr int cDT = 768;
constexpr int cK1 = cDT + cDI;   // 1024
constexpr int cK2 = cDU + cDI;   // 384
}

// issue one per-lane 16B async copy global -> LDS (ASYNCcnt-tracked, no VGPR data)
__device__ __forceinline__ void async_copy_b128(const float* gsrc, const float* ldst)
{
  const unsigned lds_addr = (unsigned)(uintptr_t)ldst;   // LDS offset = addr[31:0]
  asm volatile("global_load_async_to_lds_b128 %0, %1, off"
               :: "v"(lds_addr), "v"(gsrc) : "memory");
}

// ---------------------------------------------------------------------------
// Attention over followed users: one wave per batch row.
// ---------------------------------------------------------------------------
__global__ __launch_bounds__(128)
void attn_kernel(const int* __restrict__ user_id,
                 const int* __restrict__ user_follow,
                 const int* __restrict__ follow_len,
                 const float* __restrict__ utab,
                 float* __restrict__ ue)
{
  const int wave = threadIdx.x >> 5;
  const int lane = threadIdx.x & 31;
  const int b = blockIdx.x * 4 + wave;
  __shared__ float sc[4][64];

  const int uid  = user_id[b];
  const int flen = follow_len[b];
  const float4 u4 = *(const float4*)(utab + (size_t)uid * cDU + lane * 4);

  // score[0] = dot(u, u)
  float d0 = u4.x*u4.x + u4.y*u4.y + u4.z*u4.z + u4.w*u4.w;
  #pragma unroll
  for (int o = 16; o > 0; o >>= 1) d0 += __shfl_down(d0, o);
  if (lane == 0) sc[wave][0] = d0;

  for (int j = 0; j < cF; ++j) {
    const int fid = user_follow[b * cF + j];
    const float4 f4 = *(const float4*)(utab + (size_t)fid * cDU + lane * 4);
    float t = f4.x*u4.x + f4.y*u4.y + f4.z*u4.z + f4.w*u4.w;
    #pragma unroll
    for (int o = 16; o > 0; o >>= 1) t += __shfl_down(t, o);
    if (lane == 0) sc[wave][1 + j] = (j < flen) ? t : -1e9f;
  }
  __syncthreads();

  float mx = -3.4e38f;
  for (int n = 0; n < cF + 1; ++n) mx = fmaxf(mx, sc[wave][n]);
  float den = 0.f;
  for (int n = 0; n < cF + 1; ++n) den += __expf(sc[wave][n] - mx);
  const float inv = 1.f / den;

  const float w0 = __expf(sc[wave][0] - mx) * inv;
  float ax = w0*u4.x, ay = w0*u4.y, az = w0*u4.z, aw = w0*u4.w;
  for (int j = 0; j < cF; ++j) {
    if (j >= flen) break;  // exp(-1e9 - mx) underflows to exactly 0
    const float wj = __expf(sc[wave][1 + j] - mx) * inv;
    const int fid = user_follow[b * cF + j];
    const float4 f4 = *(const float4*)(utab + (size_t)fid * cDU + lane * 4);
    ax += wj*f4.x; ay += wj*f4.y; az += wj*f4.z; aw += wj*f4.w;
  }
  float4 r4;
  if (flen > 1) { r4.x = ax; r4.y = ay; r4.z = az; r4.w = aw; }
  else          { r4 = u4; }
  *(float4*)(ue + (size_t)b * cDU + lane * 4) = r4;
}

// ---------------------------------------------------------------------------
// GEMM1: h1r[b,n] = concat(tag,itemEmb)[b,:] . fc1_w[n,:] + fc1_b[n] + itemEmb[b,n]
// One wave per 16x16 tile, f32 WMMA. K staged via DOUBLE-BUFFERED async
// global->LDS copies (16 x b128 per chunk) overlapped with the WMMA pipe.
// Also accumulates per-column sum / sumsq for BatchNorm1.
// ---------------------------------------------------------------------------
__global__ __launch_bounds__(32)
void gemm1_kernel(const float* __restrict__ tag_emb,
                  const float* __restrict__ item_tab,
                  const int* __restrict__ item_id,
                  const float* __restrict__ w1,   // [256,1024]
                  const float* __restrict__ b1,   // [256]
                  float* __restrict__ h1r,
                  float* __restrict__ sum1,
                  float* __restrict__ sumsq1)
{
  const int lane = threadIdx.x;
  const int tm = blockIdx.x, tn = blockIdx.y;
  const int half = lane >> 4, l16 = lane & 15;
  const int ldrow = lane >> 1;           // staging: row within tile
  const int ldk0  = (lane & 1) * 32;     // staging: k-half within chunk
  __shared__ float Als[2][16][68];
  __shared__ float Bls[2][16][68];
  v8f acc = {};
  const int brow = tm * 16 + ldrow;
  const int nrow = tn * 16 + ldrow;
  const int iid  = item_id[brow];

  // issue the 16 async b128 copies staging K-chunk `kb` into buffer `buf`
  auto issue = [&](int buf, int kb) {
    // chunks are 64-aligned; concat boundary is 768 -> single source per chunk
    const float* arow = (kb < cDT)
        ? tag_emb  + (size_t)brow * cDT + kb
        : item_tab + (size_t)iid  * cDI + (kb - cDT);
    const float* wrow = w1 + (size_t)nrow * cK1 + kb;
    #pragma unroll
    for (int i = 0; i < 8; ++i) {
      const int k = ldk0 + i * 4;
      async_copy_b128(arow + k, &Als[buf][ldrow][k]);
      async_copy_b128(wrow + k, &Bls[buf][ldrow][k]);
    }
  };

  issue(0, 0);
  for (int c = 0; c < cK1 / 64; ++c) {
    const int buf = c & 1;
    if (c + 1 < cK1 / 64) {
      issue(buf ^ 1, (c + 1) * 64);
      // 16 newer copies may stay in flight; the older 16 (this buffer) must land
      asm volatile("s_wait_asynccnt 0x10" ::: "memory");
    } else {
      asm volatile("s_wait_asynccnt 0x0" ::: "memory");
    }
    #pragma unroll
    for (int kk = 0; kk < 16; ++kk) {
      const int k0 = kk * 4 + half * 2;
      const v2f a  = *(const v2f*)&Als[buf][l16][k0];
      const v2f bf = *(const v2f*)&Bls[buf][l16][k0];
      acc = __builtin_amdgcn_wmma_f32_16x16x4_f32(false, a, false, bf,
                                                  (short)0, acc, false, false);
    }
    // buffer `buf` is re-written by the next iteration's issue; its ds reads complete
    // before the WMMAs above (s_wait_dscnt), which precede that issue in
    // program order (volatile asm + memory clobber pins the ordering).
  }

  const int n_g = tn * 16 + l16;
  const float bias = b1[n_g];
  float s = 0.f, sq = 0.f;
  #pragma unroll
  for (int r = 0; r < 8; ++r) {
    const int bidx = tm * 16 + half * 8 + r;
    const float v = acc[r] + bias + item_tab[(size_t)item_id[bidx] * cDI + n_g];
    h1r[(size_t)bidx * cDI + n_g] = v;
    s += v; sq += v * v;
  }
  s  += __shfl_xor(s, 16);
  sq += __shfl_xor(sq, 16);
  if (half == 0) { atomicAdd(sum1 + n_g, s); atomicAdd(sumsq1 + n_g, sq); }
}

// ---------------------------------------------------------------------------
// Fold BN stats into per-column scale/bias: y = x*scale + bias, then ReLU later.
// ---------------------------------------------------------------------------
__global__ void bn_fin_kernel(const float* __restrict__ sum,
                              const float* __restrict__ sumsq,
                              const float* __restrict__ g,
                              const float* __restrict__ be,
                              float* __restrict__ scale,
                              float* __restrict__ bias, int n)
{
  const int i = threadIdx.x;
  if (i < n) {
    const float mu  = sum[i] * (1.f / cB);
    const float var = sumsq[i] * (1.f / cB) - mu * mu;
    const float scv = g[i] * rsqrtf(var + 1e-5f);
    scale[i] = scv;
    bias[i]  = be[i] - mu * scv;
  }
}

// ---------------------------------------------------------------------------
// GEMM2: A = concat(user_embedded, relu(bn1(h1r))) [4096x384], W = fc2_w [128x384].
// BN1+ReLU fused into the LDS staging of A (VALU path); the weight tile is
// staged with async global->LDS copies. Epilogue: +fc2_b +user_embedded,
// plus BN2 stats accumulation.
// ---------------------------------------------------------------------------
__global__ __launch_bounds__(32)
void gemm2_kernel(const float* __restrict__ ue,     // [B,128]
                  const float* __restrict__ h1r,    // [B,256]
                  const float* __restrict__ scale1,
                  const float* __restrict__ bias1,
                  const float* __restrict__ w2,     // [128,384]
                  const float* __restrict__ b2,
                  float* __restrict__ h2r,
                  float* __restrict__ sum2,
                  float* __restrict__ sumsq2)
{
  const int lane = threadIdx.x;
  const int tm = blockIdx.x, tn = blockIdx.y;
  const int half = lane >> 4, l16 = lane & 15;
  const int ldrow = lane >> 1;
  const int ldk0  = (lane & 1) * 32;
  __shared__ float Als[16][68];
  __shared__ float Bls[16][68];
  v8f acc = {};
  const int brow = tm * 16 + ldrow;
  const int nrow = tn * 16 + ldrow;

  for (int kb = 0; kb < cK2; kb += 64) {
    const float* wrow = w2 + (size_t)nrow * cK2 + kb;
    // weights: raw bytes -> async copy straight into LDS
    #pragma unroll
    for (int i = 0; i < 8; ++i) {
      const int k = ldk0 + i * 4;
      async_copy_b128(wrow + k, &Bls[ldrow][k]);
    }
    // A operand: needs BN+ReLU transform for the h1r part -> VALU staging
    if (kb < cDU) {
      #pragma unroll
      for (int i = 0; i < 8; ++i) {
        const int k = ldk0 + i * 4;
        *(float4*)&Als[ldrow][k] = *(const float4*)(ue + (size_t)brow * cDU + kb + k);
      }
    } else {
      #pragma unroll
      for (int i = 0; i < 8; ++i) {
        const int k = ldk0 + i * 4;
        const int c = kb - cDU + k;
        const float4 h  = *(const float4*)(h1r + (size_t)brow * cDI + c);
        const float4 sv = *(const float4*)(scale1 + c);
        const float4 bv = *(const float4*)(bias1 + c);
        float4 r;
        r.x = fmaxf(h.x * sv.x + bv.x, 0.f);
        r.y = fmaxf(h.y * sv.y + bv.y, 0.f);
        r.z = fmaxf(h.z * sv.z + bv.z, 0.f);
        r.w = fmaxf(h.w * sv.w + bv.w, 0.f);
        *(float4*)&Als[ldrow][k] = r;
      }
    }
    asm volatile("s_wait_asynccnt 0x0" ::: "memory");
    __syncthreads();
    #pragma unroll
    for (int kk = 0; kk < 16; ++kk) {
      const int k0 = kk * 4 + half * 2;
      const v2f a  = *(const v2f*)&Als[l16][k0];
      const v2f bf = *(const v2f*)&Bls[l16][k0];
      acc = __builtin_amdgcn_wmma_f32_16x16x4_f32(false, a, false, bf,
                                                  (short)0, acc, false, false);
    }
    __syncthreads();
  }

  const int n_g = tn * 16 + l16;
  const float bias = b2[n_g];
  float s = 0.f, sq = 0.f;
  #pragma unroll
  for (int r = 0; r < 8; ++r) {
    const int bidx = tm * 16 + half * 8 + r;
    const float v = acc[r] + bias + ue[(size_t)bidx * cDU + n_g];
    h2r[(size_t)bidx * cDU + n_g] = v;
    s += v; sq += v * v;
  }
  s  += __shfl_xor(s, 16);
  sq += __shfl_xor(sq, 16);
  if (half == 0) { atomicAdd(sum2 + n_g, s); atomicAdd(sumsq2 + n_g, sq); }
}

// ---------------------------------------------------------------------------
// Output: logits[b] = relu(bn2(h2r))[b,:] . out_w[0:128] + feat[b,:] . out_w[128:131] + out_b
// One wave per row.
// ---------------------------------------------------------------------------
__global__ __launch_bounds__(256)
void out_kernel(const float* __restrict__ h2r,
                const float* __restrict__ scale2,
                const float* __restrict__ bias2,
                const float* __restrict__ feat,
                const float* __restrict__ out_w,
                const float* __restrict__ out_b,
                float* __restrict__ out)
{
  const int wave = threadIdx.x >> 5;
  const int lane = threadIdx.x & 31;
  const int b = blockIdx.x * 8 + wave;
  float s = 0.f;
  #pragma unroll
  for (int i = lane; i < cDU; i += 32) {
    const float v = fmaxf(h2r[(size_t)b * cDU + i] * scale2[i] + bias2[i], 0.f);
    s += v * out_w[i];
  }
  #pragma unroll
  for (int o = 16; o > 0; o >>= 1) s += __shfl_down(s, o);
  if (lane == 0) {
    s += feat[b*3+0]*out_w[cDU+0] + feat[b*3+1]*out_w[cDU+1] + feat[b*3+2]*out_w[cDU+2]
       + out_b[0];
    out[b] = s;
  }
}

// ---------------------------------------------------------------------------
extern "C" void kernel_launch(void* const* d_in, const int* in_sizes, int n_in,
                              void* d_out, int out_size, void* d_ws, size_t ws_size,
                              hipStream_t stream)
{
  (void)in_sizes; (void)n_in; (void)out_size; (void)ws_size;
  const int*   user_id      = (const int*)d_in[0];
  const int*   item_id      = (const int*)d_in[1];
  const int*   user_follow  = (const int*)d_in[2];
  const int*   follow_len   = (const int*)d_in[3];
  const float* feature_item = (const float*)d_in[4];
  const float* tag_emb      = (const float*)d_in[5];
  const float* utab         = (const float*)d_in[6];
  const float* itab         = (const float*)d_in[7];
  const float* fc1_w        = (const float*)d_in[8];
  const float* fc1_b        = (const float*)d_in[9];
  const float* fc2_w        = (const float*)d_in[10];
  const float* fc2_b        = (const float*)d_in[11];
  const float* out_w        = (const float*)d_in[12];
  const float* out_b        = (const float*)d_in[13];
  const float* bn1_g        = (const float*)d_in[14];
  const float* bn1_b        = (const float*)d_in[15];
  const float* bn2_g        = (const float*)d_in[16];
  const float* bn2_b        = (const float*)d_in[17];

  float* wsf    = (float*)d_ws;
  float* ue     = wsf;                          // 4096*128
  float* h1r    = ue     + (size_t)cB * cDU;    // 4096*256
  float* h2r    = h1r    + (size_t)cB * cDI;    // 4096*128
  float* sum1   = h2r    + (size_t)cB * cDU;    // 256
  float* sumsq1 = sum1   + cDI;                 // 256
  float* sum2   = sumsq1 + cDI;                 // 128
  float* sumsq2 = sum2   + cDU;                 // 128
  float* scale1 = sumsq2 + cDU;                 // 256
  float* bias1  = scale1 + cDI;                 // 256
  float* scale2 = bias1  + cDI;                 // 128
  float* bias2  = scale2 + cDU;                 // 128

  // zero the BN stat accumulators every call (sum1..sumsq2 are contiguous)
  hipMemsetAsync(sum1, 0, (size_t)(2 * cDI + 2 * cDU) * sizeof(float), stream);

  attn_kernel<<<cB / 4, 128, 0, stream>>>(user_id, user_follow, follow_len, utab, ue);
  gemm1_kernel<<<dim3(cB / 16, cDI / 16), 32, 0, stream>>>(
      tag_emb, itab, item_id, fc1_w, fc1_b, h1r, sum1, sumsq1);
  bn_fin_kernel<<<1, 256, 0, stream>>>(sum1, sumsq1, bn1_g, bn1_b, scale1, bias1, cDI);
  gemm2_kernel<<<dim3(cB / 16, cDU / 16), 32, 0, stream>>>(
      ue, h1r, scale1, bias1, fc2_w, fc2_b, h2r, sum2, sumsq2);
  bn_fin_kernel<<<1, 128, 0, stream>>>(sum2, sumsq2, bn2_g, bn2_b, scale2, bias2, cDU);
  out_kernel<<<cB / 8, 256, 0, stream>>>(h2r, scale2, bias2, feature_item, out_w, out_b,
                                         (float*)d_out);
}